// Aggr_80977313399672
// MI455X (gfx1250) — compile-verified
//
#include <hip/hip_runtime.h>

// ---------------------------------------------------------------------------
// 8-layer GCN propagation (feature dim = 1) on a 1M-node / 16M-edge graph.
//
// Roofline (MI455X): ~0.26 GFLOP vs ~1.2 GB logical traffic -> pure memory /
// atomic bound. src+dst (128 MB int32) + node arrays (~20 MB) fit in the
// 192 MB L2, so layers 2..8 run entirely out of L2. The optimal path is
// 128-bit streaming loads of edge indices + one global_atomic_add_f32 per
// edge + wave32 shuffle reductions. WMMA is not applicable (no K dimension).
// ---------------------------------------------------------------------------

#ifndef GCN_N_NODES
#define GCN_N_NODES 1000000
#endif
#ifndef GCN_N_EDGES
#define GCN_N_EDGES 16000000
#endif
#define GCN_LAYERS 8
#define TB 256

// ws layout (floats): [0,N) norm_src  [N,2N) norm_dst  [2N,3N) hs  [3N,4N) agg

__global__ __launch_bounds__(TB) void gcn_init_kernel(float* __restrict__ norm_src,
                                                      float* __restrict__ norm_dst,
                                                      float* __restrict__ agg,
                                                      float* __restrict__ out) {
    int i = blockIdx.x * TB + threadIdx.x;
    if (i < GCN_N_NODES) {
        norm_src[i] = 0.0f;
        norm_dst[i] = 0.0f;
        agg[i]      = 0.0f;
    }
    if (blockIdx.x == 0 && threadIdx.x < GCN_LAYERS) out[threadIdx.x] = 0.0f;
}

// Both degree histograms in one pass over the edge list (int4 = 4 edges/thread,
// 128-bit coalesced loads). Float atomics are exact for counts < 2^24.
__global__ __launch_bounds__(TB) void gcn_degree_kernel(const int4* __restrict__ src4,
                                                        const int4* __restrict__ dst4,
                                                        float* __restrict__ out_deg,
                                                        float* __restrict__ in_deg) {
    int i = blockIdx.x * TB + threadIdx.x;
    if (i >= GCN_N_EDGES / 4) return;
    int4 s = src4[i];
    int4 d = dst4[i];
    atomicAdd(&out_deg[s.x], 1.0f);
    atomicAdd(&out_deg[s.y], 1.0f);
    atomicAdd(&out_deg[s.z], 1.0f);
    atomicAdd(&out_deg[s.w], 1.0f);
    atomicAdd(&in_deg[d.x], 1.0f);
    atomicAdd(&in_deg[d.y], 1.0f);
    atomicAdd(&in_deg[d.z], 1.0f);
    atomicAdd(&in_deg[d.w], 1.0f);
}

// deg -> clamp(.,1)^-1/2 in place; pre-scale hs = h * norm_src for layer 0.
__global__ __launch_bounds__(TB) void gcn_norm_kernel(const float* __restrict__ h,
                                                      float* __restrict__ norm_src,
                                                      float* __restrict__ norm_dst,
                                                      float* __restrict__ hs) {
    int i = blockIdx.x * TB + threadIdx.x;
    if (i >= GCN_N_NODES) return;
    float a = norm_src[i]; a = (a < 1.0f) ? 1.0f : a;
    float b = norm_dst[i]; b = (b < 1.0f) ? 1.0f : b;
    float ra = 1.0f / sqrtf(a);   // one-time cost: favor accuracy over v_rsq
    float rb = 1.0f / sqrtf(b);
    norm_src[i] = ra;
    norm_dst[i] = rb;
    hs[i] = h[i] * ra;
}

// The hot kernel: one gather + one L2 atomic per edge. Streams indices with
// B128 loads and prefetches the streams one page ahead (global_prefetch_b8,
// speculative -> tail overrun is silently dropped per CDNA5 ISA).
__global__ __launch_bounds__(TB) void gcn_edge_kernel(const int4* __restrict__ src4,
                                                      const int4* __restrict__ dst4,
                                                      const float* __restrict__ hs,
                                                      float* __restrict__ agg) {
    int i = blockIdx.x * TB + threadIdx.x;
    if (i >= GCN_N_EDGES / 4) return;
    __builtin_prefetch((const char*)(src4 + i) + 4096, 0, 0);
    __builtin_prefetch((const char*)(dst4 + i) + 4096, 0, 0);
    int4 s = src4[i];
    int4 d = dst4[i];
    float vx = hs[s.x];
    float vy = hs[s.y];
    float vz = hs[s.z];
    float vw = hs[s.w];
    atomicAdd(&agg[d.x], vx);
    atomicAdd(&agg[d.y], vy);
    atomicAdd(&agg[d.z], vz);
    atomicAdd(&agg[d.w], vw);
}

// Fused per-layer node update:
//   h      = agg * norm_dst        (the layer output)
//   hs     = h * norm_src          (pre-scaled message for the NEXT layer)
//   agg    = 0                     (cleared for the next layer, same pass)
//   out[l]+= sum(h^2)              (wave32 shuffle + LDS + 1 atomic / block)
__global__ __launch_bounds__(TB) void gcn_node_kernel(float* __restrict__ agg,
                                                      const float* __restrict__ norm_src,
                                                      const float* __restrict__ norm_dst,
                                                      float* __restrict__ hs,
                                                      float* __restrict__ out,
                                                      int layer) {
    __shared__ float red[TB / 32];
    int i = blockIdx.x * TB + threadIdx.x;
    float v = 0.0f;
    if (i < GCN_N_NODES) {
        float a = agg[i];
        agg[i] = 0.0f;
        float hn = a * norm_dst[i];
        hs[i] = hn * norm_src[i];
        v = hn * hn;
    }
    // wave32 butterfly reduction (gfx1250 is wave32-only)
    #pragma unroll
    for (int off = 16; off > 0; off >>= 1)
        v += __shfl_xor(v, off, 32);
    int lane = threadIdx.x & 31;
    int wave = threadIdx.x >> 5;
    if (lane == 0) red[wave] = v;
    __syncthreads();
    if (wave == 0) {
        float w = (lane < (TB / 32)) ? red[lane] : 0.0f;
        #pragma unroll
        for (int off = (TB / 64); off > 0; off >>= 1)
            w += __shfl_xor(w, off, 32);
        if (lane == 0) atomicAdd(&out[layer], w);
    }
}

extern "C" void kernel_launch(void* const* d_in, const int* in_sizes, int n_in,
                              void* d_out, int out_size, void* d_ws, size_t ws_size,
                              hipStream_t stream) {
    // setup_inputs() order: h [N], src [E], dst [E], n_nodes (scalar), l (scalar)
    const float* h   = (const float*)d_in[0];
    const int*   src = (const int*)d_in[1];
    const int*   dst = (const int*)d_in[2];
    float*       out = (float*)d_out;

    float* ws       = (float*)d_ws;
    float* norm_src = ws;                    //  4 MB
    float* norm_dst = ws + (size_t)GCN_N_NODES;       //  4 MB
    float* hs       = ws + 2 * (size_t)GCN_N_NODES;   //  4 MB
    float* agg      = ws + 3 * (size_t)GCN_N_NODES;   //  4 MB  (16 MB total)

    const int nblk = (GCN_N_NODES + TB - 1) / TB;        // 3907
    const int eblk = (GCN_N_EDGES / 4 + TB - 1) / TB;    // 15625

    gcn_init_kernel<<<nblk, TB, 0, stream>>>(norm_src, norm_dst, agg, out);
    gcn_degree_kernel<<<eblk, TB, 0, stream>>>((const int4*)src, (const int4*)dst,
                                               norm_src, norm_dst);
    gcn_norm_kernel<<<nblk, TB, 0, stream>>>(h, norm_src, norm_dst, hs);

    for (int l = 0; l < GCN_LAYERS; ++l) {
        gcn_edge_kernel<<<eblk, TB, 0, stream>>>((const int4*)src, (const int4*)dst,
                                                 hs, agg);
        gcn_node_kernel<<<nblk, TB, 0, stream>>>(agg, norm_src, norm_dst, hs, out, l);
    }
}